// MaskedBalancedBCELoss_18141941858632
// MI455X (gfx1250) — compile-verified
//
#include <hip/hip_runtime.h>

// ---------------------------------------------------------------------------
// MaskedBalancedBCELoss for MI455X (gfx1250, wave32).
//
// Strategy (memory-bound: 157MB in @ 23.3TB/s ~ 6.7us floor):
//   K1  : stream pred/gt/mask (non-temporal), compute BCE loss,
//         write negLoss (52MB -> stays in 192MB L2), 4096-bin radix hist of
//         float bits[31:20], block-reduce scalars via V_WMMA_F32_16X16X4_F32.
//   K2  : compute k = min(neg_total, 3*pos_count) (0 if pos_count==0).
//   scan/refine x3 : exact 32-bit radix-select (12+12+8 bits) of the k-th
//         largest negative loss -> exact threshold pattern T + residual need.
//   K7  : sum negLoss where bits > T (L2-resident pass), WMMA block reduce.
//   K8  : result = (pos_sum + sum_above + need*value(T)) / (pos_cnt + k + eps)
// ---------------------------------------------------------------------------

typedef __attribute__((ext_vector_type(2))) float    v2f;
typedef __attribute__((ext_vector_type(4))) float    v4f;
typedef __attribute__((ext_vector_type(8))) float    v8f;
typedef __attribute__((ext_vector_type(4))) unsigned v4u;

#define TPB     256
#define NBLK    1024
#define H1_BINS 4096
#define H3_BINS 256

// sel[] layout: 0:k  1:B1  2:need1  3:B2  4:need2  5:B3  6:need3
// accF[] layout: 0:posSum  1:posCnt  2:negCnt  3:sumAboveT  (4..7 pad)

// Cross-block reduction: intra-wave shfl tree, then ONE pair of chained
// v_wmma_f32_16x16x4_f32 ops sums the 8 wave partials for up to 16 quantities
// (A = ones(16x4) => D[:,n] = sum_k B[k,n]; slot->row mapping is irrelevant,
// only column = lane%16 matters).
__device__ __forceinline__ void wmma_block_reduce(float a0, float a1, float a2,
                                                  float* dst, int nq) {
#pragma unroll
  for (int off = 16; off >= 1; off >>= 1) {
    a0 += __shfl_xor(a0, off, 32);
    a1 += __shfl_xor(a1, off, 32);
    a2 += __shfl_xor(a2, off, 32);
  }
  __shared__ float wp[8][16];
  const int lane = (int)(threadIdx.x & 31u);
  const int w    = (int)(threadIdx.x >> 5);
  if (lane < 16) wp[w][lane] = 0.0f;
  if (lane == 0) { wp[w][0] = a0; wp[w][1] = a1; wp[w][2] = a2; }
  __syncthreads();
  if (w == 0) {  // wave-uniform branch: EXEC = all ones inside (WMMA req.)
    const int q = lane & 15;
    const int h = lane >> 4;
    v2f ones; ones.x = 1.0f; ones.y = 1.0f;
    v2f b0, b1;
    b0.x = wp[0 + h][q]; b0.y = wp[2 + h][q];   // waves 0..3
    b1.x = wp[4 + h][q]; b1.y = wp[6 + h][q];   // waves 4..7
    v8f c = {0.f, 0.f, 0.f, 0.f, 0.f, 0.f, 0.f, 0.f};
    c = __builtin_amdgcn_wmma_f32_16x16x4_f32(false, ones, false, b0,
                                              (short)0, c, false, false);
    c = __builtin_amdgcn_wmma_f32_16x16x4_f32(false, ones, false, b1,
                                              (short)0, c, false, false);
    if (lane < nq) atomicAdd(&dst[lane], c[0]);  // D row M=0, col=lane
  }
}

__device__ __forceinline__ float procElem(float p, float g, float m,
                                          float& posSum, float& posCnt,
                                          float& negCnt, unsigned* hcnt) {
  float lp   = fmaxf(logf(p),      -100.0f);   // torch BCE log clamp
  float l1   = fmaxf(log1pf(-p),   -100.0f);
  float loss = -(g * lp + (1.0f - g) * l1);
  float pw = g * m;
  float nw = (1.0f - g) * m;
  posCnt += pw;
  negCnt += nw;
  posSum  = fmaf(loss, pw, posSum);
  float nl = loss * nw;                         // >= 0; 0 iff not valid negative
  atomicAdd(&hcnt[__float_as_uint(nl) >> 20], 1u);
  return nl;
}

__global__ void __launch_bounds__(TPB)
k_init(float* accF, unsigned* sel, unsigned* h1, unsigned* h2, unsigned* h3) {
  const int tid    = (int)(blockIdx.x * blockDim.x + threadIdx.x);
  const int stride = (int)(gridDim.x * blockDim.x);
  for (int i = tid; i < H1_BINS; i += stride) h1[i] = 0u;
  for (int i = tid; i < H1_BINS; i += stride) h2[i] = 0u;
  for (int i = tid; i < H3_BINS; i += stride) h3[i] = 0u;
  if (tid < 8) { accF[tid] = 0.0f; sel[tid] = 0u; }
}

__global__ void __launch_bounds__(TPB)
k_main(const float* __restrict__ pred, const float* __restrict__ gt,
       const float* __restrict__ mask, float* __restrict__ negLoss,
       float* __restrict__ accF, unsigned* __restrict__ hist1, int n) {
  __shared__ unsigned hcnt[H1_BINS];
  for (int i = (int)threadIdx.x; i < H1_BINS; i += TPB) hcnt[i] = 0u;
  __syncthreads();

  float posSum = 0.f, posCnt = 0.f, negCnt = 0.f;
  const int n4     = n >> 2;
  const int stride = (int)(gridDim.x * TPB);
  const v4f* p4 = (const v4f*)pred;
  const v4f* g4 = (const v4f*)gt;
  const v4f* m4 = (const v4f*)mask;
  v4f*       o4 = (v4f*)negLoss;

  for (int i = (int)(blockIdx.x * TPB + threadIdx.x); i < n4; i += stride) {
    // Non-temporal: streamed once, keep L2 free for the negLoss workspace.
    v4f p = __builtin_nontemporal_load(p4 + i);
    v4f g = __builtin_nontemporal_load(g4 + i);
    v4f m = __builtin_nontemporal_load(m4 + i);
    v4f nl;
    nl[0] = procElem(p[0], g[0], m[0], posSum, posCnt, negCnt, hcnt);
    nl[1] = procElem(p[1], g[1], m[1], posSum, posCnt, negCnt, hcnt);
    nl[2] = procElem(p[2], g[2], m[2], posSum, posCnt, negCnt, hcnt);
    nl[3] = procElem(p[3], g[3], m[3], posSum, posCnt, negCnt, hcnt);
    o4[i] = nl;                                  // regular store -> L2 resident
  }
  const int base = n4 << 2;
  const int tid  = (int)(blockIdx.x * TPB + threadIdx.x);
  if (tid < n - base) {
    const int i = base + tid;
    negLoss[i] = procElem(pred[i], gt[i], mask[i], posSum, posCnt, negCnt, hcnt);
  }
  __syncthreads();
  for (int i = (int)threadIdx.x; i < H1_BINS; i += TPB) {
    const unsigned c = hcnt[i];
    if (c) atomicAdd(&hist1[i], c);
  }
  wmma_block_reduce(posSum, posCnt, negCnt, accF, 3);
}

__global__ void k_compute_k(const float* __restrict__ accF,
                            unsigned* __restrict__ sel) {
  if (threadIdx.x == 0 && blockIdx.x == 0) {
    const unsigned pc = (unsigned)accF[1];   // exact: integer-valued f32 sums
    const unsigned nt = (unsigned)accF[2];
    const unsigned k3 = 3u * pc;             // NEG_RATIO = 3
    sel[0] = (pc == 0u) ? 0u : (nt < k3 ? nt : k3);  // FALLBACK_NEG = 0
  }
}

// Find bin B with count(bins > B) < k <= count(bins >= B); need = k - above.
__global__ void __launch_bounds__(TPB)
k_scan(const unsigned* __restrict__ hist, int nb,
       const unsigned* __restrict__ kPtr,
       unsigned* __restrict__ outB, unsigned* __restrict__ outNeed) {
  __shared__ unsigned sbuf[TPB];
  const unsigned k  = *kPtr;
  const int t       = (int)threadIdx.x;
  const int per     = nb / TPB;
  unsigned local[16];
  unsigned s = 0u;
  for (int j = 0; j < per; ++j) { local[j] = hist[t * per + j]; s += local[j]; }
  const unsigned chunkSelf = s;
  unsigned v = s;                              // inclusive suffix sum
  for (int off = 1; off < TPB; off <<= 1) {
    sbuf[t] = v;
    __syncthreads();
    const unsigned add = (t + off < TPB) ? sbuf[t + off] : 0u;
    __syncthreads();
    v += add;
  }
  unsigned cum = v - chunkSelf;                // count strictly above this chunk
  for (int j = per - 1; j >= 0; --j) {
    const unsigned c = local[j];
    if (cum < k && cum + c >= k) {             // unique crossing bin (k >= 1)
      *outB    = (unsigned)(t * per + j);
      *outNeed = k - cum;
    }
    cum += c;
  }
  if (k == 0u && t == 0) { *outB = (unsigned)(nb - 1); *outNeed = 0u; }
}

__global__ void __launch_bounds__(TPB)
k_refine2(const unsigned* __restrict__ bits, int n,
          const unsigned* __restrict__ sel, unsigned* __restrict__ hist2) {
  __shared__ unsigned hcnt[H1_BINS];
  for (int i = (int)threadIdx.x; i < H1_BINS; i += TPB) hcnt[i] = 0u;
  __syncthreads();
  const unsigned B1 = sel[1];
  const int n4      = n >> 2;
  const int stride  = (int)(gridDim.x * TPB);
  const v4u* b4 = (const v4u*)bits;
  for (int i = (int)(blockIdx.x * TPB + threadIdx.x); i < n4; i += stride) {
    v4u b = b4[i];
#pragma unroll
    for (int c = 0; c < 4; ++c) {
      const unsigned x = b[c];
      if ((x >> 20) == B1) atomicAdd(&hcnt[(x >> 8) & 0xFFFu], 1u);
    }
  }
  const int base = n4 << 2;
  const int tid  = (int)(blockIdx.x * TPB + threadIdx.x);
  if (tid < n - base) {
    const unsigned x = bits[base + tid];
    if ((x >> 20) == B1) atomicAdd(&hcnt[(x >> 8) & 0xFFFu], 1u);
  }
  __syncthreads();
  for (int i = (int)threadIdx.x; i < H1_BINS; i += TPB) {
    const unsigned c = hcnt[i];
    if (c) atomicAdd(&hist2[i], c);
  }
}

__global__ void __launch_bounds__(TPB)
k_refine3(const unsigned* __restrict__ bits, int n,
          const unsigned* __restrict__ sel, unsigned* __restrict__ hist3) {
  __shared__ unsigned hcnt[H3_BINS];
  for (int i = (int)threadIdx.x; i < H3_BINS; i += TPB) hcnt[i] = 0u;
  __syncthreads();
  const unsigned P2 = (sel[1] << 12) | sel[3];   // bits[31:8] prefix
  const int n4      = n >> 2;
  const int stride  = (int)(gridDim.x * TPB);
  const v4u* b4 = (const v4u*)bits;
  for (int i = (int)(blockIdx.x * TPB + threadIdx.x); i < n4; i += stride) {
    v4u b = b4[i];
#pragma unroll
    for (int c = 0; c < 4; ++c) {
      const unsigned x = b[c];
      if ((x >> 8) == P2) atomicAdd(&hcnt[x & 0xFFu], 1u);
    }
  }
  const int base = n4 << 2;
  const int tid  = (int)(blockIdx.x * TPB + threadIdx.x);
  if (tid < n - base) {
    const unsigned x = bits[base + tid];
    if ((x >> 8) == P2) atomicAdd(&hcnt[x & 0xFFu], 1u);
  }
  __syncthreads();
  for (int i = (int)threadIdx.x; i < H3_BINS; i += TPB) {
    const unsigned c = hcnt[i];
    if (c) atomicAdd(&hist3[i], c);
  }
}

__global__ void __launch_bounds__(TPB)
k_sum_above(const float* __restrict__ negLoss, int n,
            const unsigned* __restrict__ sel, float* __restrict__ accF) {
  const unsigned T = (sel[1] << 20) | (sel[3] << 8) | sel[5];
  float s = 0.f;
  const int n4     = n >> 2;
  const int stride = (int)(gridDim.x * TPB);
  const v4f* a4 = (const v4f*)negLoss;
  for (int i = (int)(blockIdx.x * TPB + threadIdx.x); i < n4; i += stride) {
    v4f v = a4[i];
#pragma unroll
    for (int c = 0; c < 4; ++c)
      if (__float_as_uint(v[c]) > T) s += v[c];
  }
  const int base = n4 << 2;
  const int tid  = (int)(blockIdx.x * TPB + threadIdx.x);
  if (tid < n - base) {
    const float v = negLoss[base + tid];
    if (__float_as_uint(v) > T) s += v;
  }
  wmma_block_reduce(s, 0.f, 0.f, accF + 3, 1);
}

__global__ void k_finalize(const float* __restrict__ accF,
                           const unsigned* __restrict__ sel,
                           float* __restrict__ out) {
  if (threadIdx.x == 0 && blockIdx.x == 0) {
    const float    posSum = accF[0];
    const float    pc     = accF[1];
    const unsigned k      = sel[0];
    const unsigned need3  = sel[6];
    const unsigned T      = (sel[1] << 20) | (sel[3] << 8) | sel[5];
    float negSum = accF[3];
    if (need3) negSum += (float)need3 * __uint_as_float(T);  // guard k==0 NaN
    out[0] = (posSum + negSum) / (pc + (float)k + 1e-6f);
  }
}

extern "C" void kernel_launch(void* const* d_in, const int* in_sizes, int n_in,
                              void* d_out, int out_size, void* d_ws, size_t ws_size,
                              hipStream_t stream) {
  const float* pred = (const float*)d_in[0];
  const float* gt   = (const float*)d_in[1];
  const float* mask = (const float*)d_in[2];
  const int n = in_sizes[0];

  char* ws = (char*)d_ws;
  size_t negBytes = ((size_t)n * 4 + 255) & ~(size_t)255;
  float*    negLoss = (float*)ws;                       // n floats (~52MB)
  float*    accF    = (float*)(ws + negBytes);          // 8 f32
  unsigned* sel     = (unsigned*)(ws + negBytes + 64);  // 8 u32
  unsigned* h1      = (unsigned*)(ws + negBytes + 128); // 4096 u32
  unsigned* h2      = h1 + H1_BINS;                     // 4096 u32
  unsigned* h3      = h2 + H1_BINS;                     // 256 u32

  k_init     <<<34,   TPB, 0, stream>>>(accF, sel, h1, h2, h3);
  k_main     <<<NBLK, TPB, 0, stream>>>(pred, gt, mask, negLoss, accF, h1, n);
  k_compute_k<<<1,    32,  0, stream>>>(accF, sel);
  k_scan     <<<1,    TPB, 0, stream>>>(h1, H1_BINS, sel + 0, sel + 1, sel + 2);
  k_refine2  <<<NBLK, TPB, 0, stream>>>((const unsigned*)negLoss, n, sel, h2);
  k_scan     <<<1,    TPB, 0, stream>>>(h2, H1_BINS, sel + 2, sel + 3, sel + 4);
  k_refine3  <<<NBLK, TPB, 0, stream>>>((const unsigned*)negLoss, n, sel, h3);
  k_scan     <<<1,    TPB, 0, stream>>>(h3, H3_BINS, sel + 4, sel + 5, sel + 6);
  k_sum_above<<<NBLK, TPB, 0, stream>>>(negLoss, n, sel, accF);
  k_finalize <<<1,    32,  0, stream>>>(accF, sel, (float*)d_out);
}